// PlasticLinear_70463233458699
// MI455X (gfx1250) — compile-verified
//
#include <hip/hip_runtime.h>
#include <hip/hip_bf16.h>

typedef __attribute__((ext_vector_type(2))) float v2f;
typedef __attribute__((ext_vector_type(4))) float v4f;
typedef __attribute__((ext_vector_type(8))) float v8f;

#define IN_F  1024
#define OUT_F 1024
#define HID   512
#define BSZ   128

// ---------------------------------------------------------------------------
// Kernel 1: modulator phi[b] = sigmoid( sum_h relu(x.W1^T + b1) * w2 + b2 )
// One block (256 threads) per batch row. ~2MB weight traffic total: trivial.
// ---------------------------------------------------------------------------
__global__ void __launch_bounds__(256)
phi_kernel(const float* __restrict__ x,
           const float* __restrict__ w1,
           const float* __restrict__ b1,
           const float* __restrict__ w2,
           const float* __restrict__ b2,
           float* __restrict__ phi) {
  __shared__ __align__(16) float xs[IN_F];
  __shared__ float red[256];
  const int b = blockIdx.x;
  const int t = threadIdx.x;

  const v4f* xr = (const v4f*)(x + (size_t)b * IN_F);
  ((v4f*)xs)[t] = xr[t];                    // 256 threads * 4 floats = 1024
  __syncthreads();

  float partial = 0.f;
  for (int h = t; h < HID; h += 256) {      // 2 hidden units per thread
    const float* wr = w1 + (size_t)h * IN_F;
    float acc = b1[h];
#pragma unroll 8
    for (int i = 0; i < IN_F; ++i) acc = fmaf(wr[i], xs[i], acc);
    acc = fmaxf(acc, 0.f);                  // ReLU
    partial = fmaf(acc, w2[h], partial);
  }
  red[t] = partial;
  __syncthreads();
  for (int s = 128; s > 0; s >>= 1) {
    if (t < s) red[t] += red[t + s];
    __syncthreads();
  }
  if (t == 0) {
    float z = red[0] + b2[0];
    phi[b] = 1.f / (1.f + __expf(-z));      // sigmoid
  }
}

// ---------------------------------------------------------------------------
// Kernel 2: y_base = x @ W^T + bias, exact f32 via V_WMMA_F32_16X16X4_F32.
// One wave (32 lanes) per 16x16 output tile; 512 tiles; K-loop of 256 steps.
// A tile (16x4 of x):  lane m=l&15, K-pair base 2*(l>=16) -> contiguous float2.
// B tile (4x16 of W^T): lane n=l&15, same K-pair base     -> contiguous float2
//   from weight row n (weight is [OUT,IN] row-major).
// D: VGPR j holds row b0+j+8*(l>=16), col n0+(l&15).
// ---------------------------------------------------------------------------
__global__ void __launch_bounds__(128)
ybase_wmma_kernel(const float* __restrict__ x,
                  const float* __restrict__ w,
                  const float* __restrict__ bias,
                  float* __restrict__ ybase) {
  const int wave = threadIdx.x >> 5;
  const int lane = threadIdx.x & 31;
  const int tile = blockIdx.x * 4 + wave;   // 0..511
  const int b0 = (tile >> 6) * 16;          // batch-tile (8 of them)
  const int n0 = (tile & 63) * 16;          // out-tile   (64 of them)
  const int half = lane >> 4;               // 0: K=0,1   1: K=2,3
  const int l15  = lane & 15;

  const float* arow = x + (size_t)(b0 + l15) * IN_F + 2 * half;
  const float* brow = w + (size_t)(n0 + l15) * IN_F + 2 * half;

  v8f c = {};
  for (int k = 0; k < IN_F; k += 4) {
    v2f a  = *(const v2f*)(arow + k);
    v2f bb = *(const v2f*)(brow + k);
    // (neg_a, A, neg_b, B, c_mod, C, reuse_a, reuse_b)
    c = __builtin_amdgcn_wmma_f32_16x16x4_f32(false, a, false, bb,
                                              (short)0, c, false, false);
  }

  const int col = n0 + l15;
  const float bc = bias[col];
#pragma unroll
  for (int j = 0; j < 8; ++j) {
    const int row = b0 + j + half * 8;
    ybase[(size_t)row * OUT_F + col] = c[j] + bc;
  }
}

// ---------------------------------------------------------------------------
// Kernel 3: single-pass fused stream over fast_state (the 536MB tensor).
// Block = (b, 8 consecutive o); one wave per (b,o) row.
//   - x[b,:] staged once in LDS per block (shared by 8 waves)
//   - each lane loads 8 float4 (32 floats) of the fast row -> registers
//   - dot with x, wave32 xor-reduce, compute y
//   - rewrite new_fast from the SAME registers: fast read exactly once.
// fast_state is 536MB streamed once (> 192MB L2): use non-temporal hints
// (TH_NT on global_load_b128/global_store_b128) so the stream doesn't evict
// the genuinely reused data (x rows, ybase, alpha, phi) from L2.
// Traffic = 536MB read + 536MB write ~= 46us at 23.3 TB/s (HBM-bound floor).
// ---------------------------------------------------------------------------
__global__ void __launch_bounds__(256)
stream_update_kernel(const float* __restrict__ x,
                     const float* __restrict__ fast,
                     const float* __restrict__ alpha,
                     const float* __restrict__ phi,
                     const float* __restrict__ ybase,
                     const float* __restrict__ eta_base_p,
                     const float* __restrict__ decay_p,
                     float* __restrict__ y_out,
                     float* __restrict__ fast_out) {
  __shared__ __align__(16) float xs[IN_F];
  const int b  = blockIdx.y;
  const int o0 = blockIdx.x * 8;
  const int t  = threadIdx.x;

  const v4f* xr = (const v4f*)(x + (size_t)b * IN_F);
  ((v4f*)xs)[t] = xr[t];
  __syncthreads();

  const int wv = t >> 5;
  const int lane = t & 31;
  const int o = o0 + wv;

  const size_t rowoff = ((size_t)b * OUT_F + o) * IN_F;
  const float* fr = fast + rowoff;

  v4f fv[8];
  float acc = 0.f;
#pragma unroll
  for (int i = 0; i < 8; ++i) {
    const int idx = i * 128 + lane * 4;       // contiguous 512B per chunk/wave
    fv[i] = __builtin_nontemporal_load((const v4f*)(fr + idx));
    const v4f xv = *(const v4f*)(xs + idx);
    acc = fmaf(fv[i].x, xv.x, acc);
    acc = fmaf(fv[i].y, xv.y, acc);
    acc = fmaf(fv[i].z, xv.z, acc);
    acc = fmaf(fv[i].w, xv.w, acc);
  }
  // wave32 butterfly reduce: every lane ends with the full dot product
#pragma unroll
  for (int m = 16; m > 0; m >>= 1) acc += __shfl_xor(acc, m, 32);

  const float ph = phi[b];
  const float yv = ybase[(size_t)b * OUT_F + o] + acc * alpha[o] * ph;
  if (lane == 0) y_out[(size_t)b * OUT_F + o] = yv;

  const float coef = eta_base_p[0] * ph * yv;   // eta * y
  const float dec  = decay_p[0];
  float* fo = fast_out + rowoff;
#pragma unroll
  for (int i = 0; i < 8; ++i) {
    const int idx = i * 128 + lane * 4;
    const v4f xv = *(const v4f*)(xs + idx);
    v4f nv;
    nv.x = fmaf(dec, fv[i].x, coef * xv.x);
    nv.y = fmaf(dec, fv[i].y, coef * xv.y);
    nv.z = fmaf(dec, fv[i].z, coef * xv.z);
    nv.w = fmaf(dec, fv[i].w, coef * xv.w);
    __builtin_nontemporal_store(nv, (v4f*)(fo + idx));
  }
}

// ---------------------------------------------------------------------------
extern "C" void kernel_launch(void* const* d_in, const int* in_sizes, int n_in,
                              void* d_out, int out_size, void* d_ws, size_t ws_size,
                              hipStream_t stream) {
  const float* x        = (const float*)d_in[0];
  const float* fast     = (const float*)d_in[1];
  const float* weight   = (const float*)d_in[2];
  const float* bias     = (const float*)d_in[3];
  const float* alpha    = (const float*)d_in[4];
  const float* mod_w1   = (const float*)d_in[5];
  const float* mod_b1   = (const float*)d_in[6];
  const float* mod_w2   = (const float*)d_in[7];
  const float* mod_b2   = (const float*)d_in[8];
  const float* eta_base = (const float*)d_in[9];
  const float* decay    = (const float*)d_in[10];

  float* y_out    = (float*)d_out;                          // [128,1024]
  float* fast_out = (float*)d_out + (size_t)BSZ * OUT_F;    // [128,1024,1024]

  // workspace: phi (128 f32) | ybase (128*1024 f32)
  float* phi   = (float*)d_ws;
  float* ybase = (float*)d_ws + 256;                        // keep 1KB pad/align

  phi_kernel<<<BSZ, 256, 0, stream>>>(x, mod_w1, mod_b1, mod_w2, mod_b2, phi);

  // 512 tiles (8 batch-tiles x 64 out-tiles), 4 waves/block -> 128 blocks
  ybase_wmma_kernel<<<128, 128, 0, stream>>>(x, weight, bias, ybase);

  stream_update_kernel<<<dim3(OUT_F / 8, BSZ), 256, 0, stream>>>(
      x, fast, alpha, phi, ybase, eta_base, decay, y_out, fast_out);
}